// SimpleNonLocal_43636867727583
// MI455X (gfx1250) — compile-verified
//
#include <hip/hip_runtime.h>

// ---------------------------------------------------------------------------
// SimpleNonLocal (patched low-rank non-local block), fp32, gfx1250 WMMA.
//
// Math reduction (per patch p, F=[C=256, n=3136], d=32):
//   Bn = W_B F   Dn = W_D F                    (stage A, WMMA)
//   S  = Bn^T Dn ; sumB, sumD                  (stage B, WMMA + fixed-order sums)
//   T  = S/n - (sumB sumD^T)/n^2 ; v = (sumB/n)^T T
//   M  = W_phi T^T ; w0 = W_phi v              (stage B, scalar tail, tiny)
//   out = F + M Bn^T - w0 1^T                  (stage C, WMMA, fused residual)
// ---------------------------------------------------------------------------

typedef float v2f __attribute__((ext_vector_type(2)));
typedef float v8f __attribute__((ext_vector_type(8)));

#define C_DIM   256
#define IMG     168
#define HW      28224        // 168*168
#define PH      56
#define NPOS    3136         // 56*56
#define DD      32
#define NPATCH  72
#define SPLIT   7
#define POS_PER_BLK 448      // NPOS / SPLIT
#define NT_PER_WAVE 7        // (POS_PER_BLK/16) / 4 waves

// workspace layout (floats)
#define BNT_OFF 0
#define DNT_OFF (NPATCH*DD*NPOS)                 // 7,225,344
#define M_OFF   (2*NPATCH*DD*NPOS)               // 14,450,688
#define W0_OFF  (M_OFF + NPATCH*C_DIM*DD)        // 15,040,512
// total 15,058,944 floats = 57.5 MiB

__device__ __forceinline__ v8f wmma4(v2f a, v2f b, v8f c) {
    // V_WMMA_F32_16X16X4_F32 : D = A(16x4) * B(4x16) + C(16x16)
    return __builtin_amdgcn_wmma_f32_16x16x4_f32(
        /*neg_a=*/false, a, /*neg_b=*/false, b,
        /*c_mod=*/(short)0, c, /*reuse_a=*/false, /*reuse_b=*/false);
}

// ---------------------------------------------------------------------------
// Stage A: BnT/DnT = [W_B; W_D] * F, stored d-major [P][32][n].
// grid = NPATCH*SPLIT, block = 128 (4 waves)
// ---------------------------------------------------------------------------
__global__ __launch_bounds__(128) void psnl_bd_kernel(
    const float* __restrict__ x, const float* __restrict__ WB,
    const float* __restrict__ WD, float* __restrict__ ws)
{
    __shared__ __align__(16) float Wlds[64 * 256];   // 64 KB: rows 0-31 W_B, 32-63 W_D
    const int tid = threadIdx.x;
    for (int idx = tid; idx < 64 * 256; idx += 128) {
        int r = idx >> 8, c = idx & 255;
        Wlds[idx] = (r < 32) ? WB[r * 256 + c] : WD[(r - 32) * 256 + c];
    }
    __syncthreads();

    const int p     = blockIdx.x / SPLIT;
    const int split = blockIdx.x % SPLIT;
    const int b  = p / 9, gy = (p / 3) % 3, gx = p % 3;
    const int lane = tid & 31, wv = tid >> 5;
    const int colN = lane & 15;
    const int ks   = (lane < 16) ? 0 : 2;
    const int mo   = (lane < 16) ? 0 : 8;

    const float* xb  = x + (size_t)b * C_DIM * HW;
    float* BnT = ws + BNT_OFF + (size_t)p * DD * NPOS;
    float* DnT = ws + DNT_OFF + (size_t)p * DD * NPOS;

    for (int j = 0; j < NT_PER_WAVE; ++j) {
        const int nt = split * POS_PER_BLK + (wv + 4 * j) * 16;
        const int i  = nt + colN;
        const int h  = gy * PH + i / PH;
        const int w  = gx * PH + i % PH;
        const float* xpix = xb + h * IMG + w;      // + c*HW for channel c

        v8f acc0 = {}, acc1 = {}, acc2 = {}, acc3 = {};
        for (int k = 0; k < C_DIM; k += 4) {
            const int kb = k + ks;
            v2f bv;                                 // B operand: F[kb..kb+1][i]
            bv.x = xpix[(size_t)kb * HW];
            bv.y = xpix[(size_t)(kb + 1) * HW];
            v2f a0 = *(const v2f*)&Wlds[( 0 + colN) * 256 + kb];
            v2f a1 = *(const v2f*)&Wlds[(16 + colN) * 256 + kb];
            v2f a2 = *(const v2f*)&Wlds[(32 + colN) * 256 + kb];
            v2f a3 = *(const v2f*)&Wlds[(48 + colN) * 256 + kb];
            acc0 = wmma4(a0, bv, acc0);
            acc1 = wmma4(a1, bv, acc1);
            acc2 = wmma4(a2, bv, acc2);
            acc3 = wmma4(a3, bv, acc3);
        }
        #pragma unroll
        for (int v = 0; v < 8; ++v) {
            BnT[( 0 + v + mo) * NPOS + i] = acc0[v];
            BnT[(16 + v + mo) * NPOS + i] = acc1[v];
            DnT[( 0 + v + mo) * NPOS + i] = acc2[v];
            DnT[(16 + v + mo) * NPOS + i] = acc3[v];
        }
    }
}

// ---------------------------------------------------------------------------
// Stage B: sums, S (WMMA), T, v, M, w0.   grid = NPATCH, block = 128 (4 waves)
// ---------------------------------------------------------------------------
__global__ __launch_bounds__(128) void psnl_t_kernel(
    const float* __restrict__ Wphi, float* __restrict__ ws)
{
    __shared__ float ps[64][2];
    __shared__ float sums[64];          // [0,32)=sumB, [32,64)=sumD
    __shared__ float Slds[32][32];
    __shared__ float Tlds[32][32];
    __shared__ float vlds[32];

    const int p = blockIdx.x;
    const int tid = threadIdx.x;
    const float* BnT = ws + BNT_OFF + (size_t)p * DD * NPOS;
    const float* DnT = ws + DNT_OFF + (size_t)p * DD * NPOS;

    // column sums of Bn and Dn (fixed-order, deterministic)
    {
        const int r = tid >> 1, part = tid & 1;
        const float* row = (r < 32) ? (BnT + r * NPOS) : (DnT + (r - 32) * NPOS);
        float s = 0.f;
        for (int i = part; i < NPOS; i += 2) s += row[i];
        ps[r][part] = s;
    }
    __syncthreads();
    if (tid < 64) sums[tid] = ps[tid][0] + ps[tid][1];
    __syncthreads();

    // S = Bn^T Dn: wave wv computes 16x16 tile (dt, et), K = positions
    const int lane = tid & 31, wv = tid >> 5;
    const int dt = wv & 1, et = wv >> 1;
    const int colN = lane & 15;
    const int ks = (lane < 16) ? 0 : 2;
    const int mo = (lane < 16) ? 0 : 8;
    const float* Arow = BnT + (dt * 16 + colN) * NPOS;   // A[m=d][k=i]
    const float* Brow = DnT + (et * 16 + colN) * NPOS;   // B[k=i][n=e]
    v8f acc = {};
    for (int k = 0; k < NPOS; k += 4) {
        v2f a  = *(const v2f*)&Arow[k + ks];
        v2f bb = *(const v2f*)&Brow[k + ks];
        acc = wmma4(a, bb, acc);
    }
    #pragma unroll
    for (int v = 0; v < 8; ++v)
        Slds[dt * 16 + v + mo][et * 16 + colN] = acc[v];
    __syncthreads();

    // T = S/n - sumB sumD^T / n^2
    const float inv_n = 1.0f / (float)NPOS;
    for (int idx = tid; idx < 1024; idx += 128) {
        int d = idx >> 5, e = idx & 31;
        Tlds[d][e] = (Slds[d][e] - sums[d] * sums[32 + e] * inv_n) * inv_n;
    }
    __syncthreads();
    if (tid < 32) {                      // v[e] = sum_d muB[d] * T[d][e]
        float s = 0.f;
        for (int d = 0; d < 32; ++d) s += (sums[d] * inv_n) * Tlds[d][tid];
        vlds[tid] = s;
    }
    __syncthreads();

    // M = W_phi T^T  [256x32], w0 = W_phi v  [256]
    float* Mout  = ws + M_OFF  + (size_t)p * C_DIM * DD;
    float* w0out = ws + W0_OFF + (size_t)p * C_DIM;
    for (int c = tid; c < C_DIM; c += 128) {
        float wreg[32];
        #pragma unroll
        for (int e = 0; e < 32; ++e) wreg[e] = Wphi[c * DD + e];
        float w0 = 0.f;
        #pragma unroll
        for (int e = 0; e < 32; ++e) w0 += wreg[e] * vlds[e];
        w0out[c] = w0;
        for (int d = 0; d < 32; ++d) {
            float m = 0.f;
            #pragma unroll
            for (int e = 0; e < 32; ++e) m += wreg[e] * Tlds[d][e];
            Mout[c * DD + d] = m;
        }
    }
}

// ---------------------------------------------------------------------------
// Stage C: out = F + M Bn^T - w0.  grid = NPATCH*SPLIT, block = 128 (4 waves)
// ---------------------------------------------------------------------------
__global__ __launch_bounds__(128) void psnl_out_kernel(
    const float* __restrict__ x, const float* __restrict__ wsc,
    float* __restrict__ out)
{
    __shared__ __align__(16) float Mlds[C_DIM * DD];   // 32 KB
    __shared__ float w0lds[C_DIM];
    const int tid = threadIdx.x;
    const int p = blockIdx.x / SPLIT, split = blockIdx.x % SPLIT;

    const float* Mg  = wsc + M_OFF  + (size_t)p * C_DIM * DD;
    const float* w0g = wsc + W0_OFF + (size_t)p * C_DIM;
    for (int idx = tid; idx < C_DIM * DD; idx += 128) Mlds[idx] = Mg[idx];
    for (int idx = tid; idx < C_DIM; idx += 128) w0lds[idx] = w0g[idx];
    __syncthreads();

    const int b = p / 9, gy = (p / 3) % 3, gx = p % 3;
    const int lane = tid & 31, wv = tid >> 5;
    const int colN = lane & 15;
    const int ks = (lane < 16) ? 0 : 2;
    const int mo = (lane < 16) ? 0 : 8;
    const float* BnT = wsc + BNT_OFF + (size_t)p * DD * NPOS;
    const float* xb = x + (size_t)b * C_DIM * HW;
    float* ob = out + (size_t)b * C_DIM * HW;

    for (int j = 0; j < NT_PER_WAVE; ++j) {
        const int nt = split * POS_PER_BLK + (wv + 4 * j) * 16;
        const int i  = nt + colN;
        const int h  = gy * PH + i / PH;
        const int w  = gx * PH + i % PH;
        const int pix = h * IMG + w;

        v2f bk[8];                      // B operand: BnT[k..k+1][i], whole K=32
        #pragma unroll
        for (int t = 0; t < 8; ++t) {
            const int kb = 4 * t + ks;
            bk[t].x = BnT[kb * NPOS + i];
            bk[t].y = BnT[(kb + 1) * NPOS + i];
        }
        for (int rt = 0; rt < 16; ++rt) {
            v8f acc = {};
            #pragma unroll
            for (int t = 0; t < 8; ++t) {
                v2f a = *(const v2f*)&Mlds[(rt * 16 + colN) * DD + 4 * t + ks];
                acc = wmma4(a, bk[t], acc);
            }
            #pragma unroll
            for (int v = 0; v < 8; ++v) {
                const int c = rt * 16 + v + mo;
                const size_t addr = (size_t)c * HW + pix;
                ob[addr] = xb[addr] + acc[v] - w0lds[c];
            }
        }
    }
}

// ---------------------------------------------------------------------------
extern "C" void kernel_launch(void* const* d_in, const int* in_sizes, int n_in,
                              void* d_out, int out_size, void* d_ws, size_t ws_size,
                              hipStream_t stream) {
    const float* x    = (const float*)d_in[0];
    const float* WB   = (const float*)d_in[1];
    const float* WD   = (const float*)d_in[2];
    const float* Wphi = (const float*)d_in[3];
    float* out = (float*)d_out;
    float* ws  = (float*)d_ws;
    (void)in_sizes; (void)n_in; (void)out_size; (void)ws_size;

    psnl_bd_kernel<<<dim3(NPATCH * SPLIT), dim3(128), 0, stream>>>(x, WB, WD, ws);
    psnl_t_kernel<<<dim3(NPATCH), dim3(128), 0, stream>>>(Wphi, ws);
    psnl_out_kernel<<<dim3(NPATCH * SPLIT), dim3(128), 0, stream>>>(x, ws, out);
}